// GroupMaxSquareLoss_20512763806265
// MI455X (gfx1250) — compile-verified
//
#include <hip/hip_runtime.h>
#include <hip/hip_bf16.h>

// Problem constants (from the reference): inputs (8, 21, 512, 512) fp32
#define N_IMG   8
#define NCH     21
#define HW      262144              // 512*512
#define OLDCL   15
#define RATIO   0.2f
#define BPI     256                 // blocks per image (pass 1)
#define TOTAL_ELEMS 44040192.0f     // 8*21*512*512

typedef float v2f __attribute__((ext_vector_type(2)));
typedef float v8f __attribute__((ext_vector_type(8)));

// ---------------------------------------------------------------------------
// Pass 1: one streaming sweep over the tensor.
// grid = (BPI, N_IMG), block = 256. Each thread handles HW/(BPI*256) = 4 pixels.
// Produces: per-image integer histogram (global atomics on uint, deterministic)
// and 7 per-image partial sums per block (no float atomics, deterministic).
// ---------------------------------------------------------------------------
__global__ __launch_bounds__(256) void gmsl_pass1(
    const float* __restrict__ x,
    unsigned int* __restrict__ gHist,   // [N_IMG][NCH]
    float* __restrict__ gPart)          // [N_IMG][BPI][7]
{
  const int n   = blockIdx.y;
  const int tid = threadIdx.x;

  __shared__ unsigned int shHist[NCH];
  __shared__ float red[256];
  __shared__ float blockS[7];

  if (tid < NCH) shHist[tid] = 0u;
  __syncthreads();

  const float* base = x + (size_t)n * NCH * HW;

  float acc[7];
#pragma unroll
  for (int j = 0; j < 7; ++j) acc[j] = 0.0f;

  const int stride = BPI * 256;
  for (int p = blockIdx.x * 256 + tid; p < HW; p += stride) {
    // prefetch next chunk of channel-0 plane (gfx1250 global_prefetch_b8)
    if (p + stride < HW) __builtin_prefetch(base + p + stride, 0, 1);

    float v[NCH];
    float m = -3.402823466e38f;
    int   am = 0;
#pragma unroll
    for (int c = 0; c < NCH; ++c) {
      v[c] = base[(size_t)c * HW + p];
      if (v[c] > m) { m = v[c]; am = c; }   // strict '>' keeps first max (jnp.argmax)
    }

    float denom = 0.0f, s0 = 0.0f;
    float e[NCH - OLDCL];
#pragma unroll
    for (int c = 0; c < NCH; ++c) {
      float ec = __expf(v[c] - m);
      denom += ec;
      if (c < OLDCL) s0 += ec;
      else           e[c - OLDCL] = ec;
    }
    const float inv = 1.0f / denom;

    const float g0 = s0 * inv;
    acc[0] += g0 * g0;
#pragma unroll
    for (int k = 0; k < NCH - OLDCL; ++k) {
      float pk = e[k] * inv;
      acc[1 + k] += pk * pk;
    }
    atomicAdd(&shHist[am], 1u);
  }
  __syncthreads();

  // tree-reduce the 7 accumulators across the block
  for (int j = 0; j < 7; ++j) {
    red[tid] = acc[j];
    __syncthreads();
    for (int off = 128; off > 0; off >>= 1) {
      if (tid < off) red[tid] += red[tid + off];
      __syncthreads();
    }
    if (tid == 0) blockS[j] = red[0];
    __syncthreads();
  }

  if (tid < 7)
    gPart[((size_t)n * BPI + blockIdx.x) * 7 + tid] = blockS[tid];
  if (tid < NCH)
    atomicAdd(&gHist[n * NCH + tid], shHist[tid]);
}

// ---------------------------------------------------------------------------
// Pass 2: single block of 256 threads.
// 1) deterministically fold the per-block partials into S[n][7]
// 2) threads 0..7 do the per-image histogram->weight math
// 3) wave 0 (32 lanes, EXEC all-ones) reduces the 168 (padded 192) products
//    S[n,c]*w[n,c] with V_WMMA_F32_16X16X4_F32, B = ones. Summing ALL 256
//    elements of D and dividing by 16 is layout-agnostic (each row-sum is
//    replicated across the 16 columns).
// ---------------------------------------------------------------------------
__global__ __launch_bounds__(256) void gmsl_pass2(
    const unsigned int* __restrict__ gHist,
    const float* __restrict__ gPart,
    float* __restrict__ out)
{
  const int t = threadIdx.x;
  __shared__ float S[N_IMG * 7];
  __shared__ float prod[192];           // 8 images * 24-slot padded rows

  if (t < N_IMG * 7) {
    const int n = t / 7, j = t % 7;
    float s = 0.0f;
    for (int b = 0; b < BPI; ++b)
      s += gPart[((size_t)n * BPI + b) * 7 + j];
    S[t] = s;
  }
  for (int i = t; i < 192; i += 256) prod[i] = 0.0f;
  __syncthreads();

  if (t < N_IMG) {
    const int n = t;
    const unsigned int* h = gHist + n * NCH;
    float h0 = 0.0f;
    for (int c = 0; c < OLDCL; ++c) h0 += (float)h[c];
    float b0 = (h0 == 0.0f) ? 1.0f : h0;     // empty folded bin -> 1
    float bt[NCH - OLDCL];
    float total = b0;
    for (int k = 0; k < NCH - OLDCL; ++k) {
      float bc = (float)h[OLDCL + k];
      bc = (bc == 0.0f) ? 1.0f : bc;         // empty bin -> 1
      bt[k] = bc;
      total += bc;                            // old bins 1..14 contribute 0
    }
    prod[n * 24 + 0] = S[n * 7 + 0] * powf(total / b0, RATIO);
    for (int k = 0; k < NCH - OLDCL; ++k)
      prod[n * 24 + OLDCL + k] = S[n * 7 + 1 + k] * powf(total / bt[k], RATIO);
  }
  __syncthreads();

  if (t < 32) {                 // wave 0: full 32-lane EXEC for WMMA
    v8f c;
#pragma unroll
    for (int i = 0; i < 8; ++i) c[i] = 0.0f;
    v2f ones;
    ones.x = 1.0f; ones.y = 1.0f;

    const int M  = t & 15;              // A row for this lane (assumed layout)
    const int K0 = (t >> 4) << 1;       // first K this lane holds

#pragma unroll
    for (int it = 0; it < 3; ++it) {    // 3 x (16x4) tiles cover 192 products
      v2f a;
      a.x = prod[it * 64 + M * 4 + K0];
      a.y = prod[it * 64 + M * 4 + K0 + 1];
      // D = A x ones + C  -> every column of D row i equals rowsum(A, i)
      c = __builtin_amdgcn_wmma_f32_16x16x4_f32(
          /*neg_a=*/false, a, /*neg_b=*/false, ones,
          /*c_mod=*/(short)0, c, /*reuse_a=*/false, /*reuse_b=*/false);
    }

    float s = 0.0f;
#pragma unroll
    for (int i = 0; i < 8; ++i) s += c[i];
#pragma unroll
    for (int off = 16; off > 0; off >>= 1) s += __shfl_xor(s, off, 32);

    if (t == 0)
      out[0] = -(s / 16.0f) / TOTAL_ELEMS;   // mean over N*C*H*W, negated
  }
}

extern "C" void kernel_launch(void* const* d_in, const int* in_sizes, int n_in,
                              void* d_out, int out_size, void* d_ws, size_t ws_size,
                              hipStream_t stream) {
  const float* x   = (const float*)d_in[0];
  float*       out = (float*)d_out;

  unsigned int* gHist = (unsigned int*)d_ws;                 // 8*21 uints
  float*        gPart = (float*)((char*)d_ws + 1024);        // 8*256*7 floats

  // zero only the histogram region (partials are fully overwritten each call)
  hipMemsetAsync(d_ws, 0, 1024, stream);

  dim3 grid(BPI, N_IMG);
  gmsl_pass1<<<grid, 256, 0, stream>>>(x, gHist, gPart);
  gmsl_pass2<<<1, 256, 0, stream>>>(gHist, gPart, out);
}